// Matcher_47949014892991
// MI455X (gfx1250) — compile-verified
//
#include <hip/hip_runtime.h>

// Problem sizes (fixed by the reference: N=1024 anchors, B=128, M=256 gt boxes)
#define N_ANCH 1024
#define BATCH  128
#define M_GT   256
#define NCHUNKS 8
#define CHUNK  (N_ANCH / NCHUNKS)   // 128 anchors per block

typedef unsigned int v4u __attribute__((ext_vector_type(4)));
typedef int          v8i __attribute__((ext_vector_type(8)));
typedef int          v4i __attribute__((ext_vector_type(4)));

// ---------------------------------------------------------------------------
// Kernel 1: one block = (batch b, anchor-chunk c); one thread = gt box m.
// TDM-load the chunk's anchors into LDS, stream IoUs with NT stores,
// keep a per-chunk argmax partial in the workspace.
// ---------------------------------------------------------------------------
__global__ __launch_bounds__(256)
void iou_partial_kernel(const float* __restrict__ anchors,
                        const float* __restrict__ gt,
                        float* __restrict__ ious,
                        float* __restrict__ wsv,
                        int*   __restrict__ wsi)
{
    __shared__ alignas(16) float sA[CHUNK * 2];   // 1 KB: (aw, ah) pairs

    const int b = blockIdx.x / NCHUNKS;
    const int c = blockIdx.x % NCHUNKS;
    const int m = threadIdx.x;

    // ---- Tensor Data Mover: DMA CHUNK*2 floats of anchors into LDS.
    // Only wave 0 issues the descriptor (TDM ignores EXEC; branch is
    // wave-uniform so waves 1..7 skip the instruction entirely).
    if (threadIdx.x < 32) {
        const unsigned long long ga =
            (unsigned long long)(anchors + (size_t)c * CHUNK * 2);
        const unsigned lds_off = (unsigned)(unsigned long long)&sA[0];

        // D# group 0 (128b): count=1 | lds_addr | global_addr[56:0] | type=2
        v4u g0;
        g0.x = 1u;                                  // count=1, user descriptor
        g0.y = lds_off;                             // lds_addr (bytes)
        g0.z = (unsigned)(ga & 0xFFFFFFFFull);      // global_addr[31:0]
        g0.w = (unsigned)((ga >> 32) & 0x1FFFFFFull) | (2u << 30); // [56:32]|type=2

        // D# group 1 (256b): data_size=4B, 1-D tile of CHUNK*2 elements
        v8i g1;
        g1[0] = 0x20000;              // workgroup_mask=0 | data_size=2 (4B)
        g1[1] = (CHUNK * 2) << 16;    // tensor_dim0[15:0] = 256 (elements)
        g1[2] = 1 << 16;              // tensor_dim0[31:16]=0 | tensor_dim1=1
        g1[3] = (CHUNK * 2) << 16;    // tile_dim0 = 256 elements
        g1[4] = 0;                    // tile_dim1=0 (unused), tile_dim2=0
        g1[5] = CHUNK * 2;            // tensor_dim0_stride = 256
        g1[6] = 0;
        g1[7] = 0;

        v4i gz4 = {0, 0, 0, 0};       // groups 2/3 unused (<=2D tensor)
        v8i gz8 = {0, 0, 0, 0, 0, 0, 0, 0};

        __builtin_amdgcn_tensor_load_to_lds(g0, g1, gz4, gz4, gz8, 0);
        __builtin_amdgcn_s_wait_tensorcnt(0);
    }
    __syncthreads();
    asm volatile("" ::: "memory");    // LDS now holds anchor data

    // ---- per-thread gt box (coalesced 16B load)
    const float4 box = reinterpret_cast<const float4*>(gt)[b * M_GT + m];
    const float gtw   = box.z - box.x;
    const float gth   = box.w - box.y;
    const float areab = gtw * gth;

    float best = -1.0f;               // IoUs are strictly > 0 here
    int   bidx = 0;
    float* __restrict__ outp =
        ious + ((size_t)b * N_ANCH + (size_t)c * CHUNK) * M_GT + m;

    #pragma unroll 8
    for (int n = 0; n < CHUNK; ++n) {
        const float2 a2 = reinterpret_cast<const float2*>(sA)[n]; // broadcast ds_load_b64
        const float aw = a2.x, ah = a2.y;
        // Center-aligned boxes: intersection == min of extents (clip is identity)
        const float iw    = fminf(aw, gtw);
        const float ih    = fminf(ah, gth);
        const float inter = iw * ih;
        const float uni   = (aw * ah + areab) - inter;
        const float iou   = inter * __builtin_amdgcn_rcpf(uni);
        __builtin_nontemporal_store(iou, outp + (size_t)n * M_GT); // write-once stream
        if (iou > best) { best = iou; bidx = c * CHUNK + n; }      // first-max ties
    }

    if (wsv) {
        const int slot = (b * M_GT + m) * NCHUNKS + c;
        wsv[slot] = best;
        wsi[slot] = bidx;
    }
}

// ---------------------------------------------------------------------------
// Kernel 2a: combine the 8 chunk partials (ascending chunk order + strict '>'
// preserves argmax first-occurrence semantics).
// ---------------------------------------------------------------------------
__global__ __launch_bounds__(256)
void reduce_ws_kernel(const float* __restrict__ wsv,
                      const int*   __restrict__ wsi,
                      float* __restrict__ matches)
{
    const int b = blockIdx.x, m = threadIdx.x;
    const int base = (b * M_GT + m) * NCHUNKS;
    float best = -1.0f;
    int   bidx = 0;
    #pragma unroll
    for (int c = 0; c < NCHUNKS; ++c) {
        const float v = wsv[base + c];
        if (v > best) { best = v; bidx = wsi[base + c]; }
    }
    matches[b * M_GT + m] = (float)bidx;
}

// ---------------------------------------------------------------------------
// Kernel 2b (fallback when ws is too small): argmax by re-reading ious.
// Loads are coalesced across m for each n.
// ---------------------------------------------------------------------------
__global__ __launch_bounds__(256)
void reduce_full_kernel(const float* __restrict__ ious,
                        float* __restrict__ matches)
{
    const int b = blockIdx.x, m = threadIdx.x;
    const float* __restrict__ p = ious + (size_t)b * N_ANCH * M_GT + m;
    float best = -1.0f;
    int   bidx = 0;
    for (int n = 0; n < N_ANCH; ++n) {
        const float v = p[(size_t)n * M_GT];
        if (v > best) { best = v; bidx = n; }
    }
    matches[b * M_GT + m] = (float)bidx;
}

// ---------------------------------------------------------------------------
extern "C" void kernel_launch(void* const* d_in, const int* in_sizes, int n_in,
                              void* d_out, int out_size, void* d_ws, size_t ws_size,
                              hipStream_t stream)
{
    const float* anchors = (const float*)d_in[0];   // (1024, 2) f32
    const float* gt      = (const float*)d_in[1];   // (128, 256, 4) f32

    float* matches = (float*)d_out;                 // first B*M elements
    float* ious    = matches + (size_t)BATCH * M_GT;// then B*N*M elements

    const size_t nslots  = (size_t)BATCH * M_GT * NCHUNKS;
    const size_t ws_need = nslots * (sizeof(float) + sizeof(int));

    if (d_ws && ws_size >= ws_need) {
        float* wsv = (float*)d_ws;
        int*   wsi = (int*)((char*)d_ws + nslots * sizeof(float));
        iou_partial_kernel<<<BATCH * NCHUNKS, 256, 0, stream>>>(
            anchors, gt, ious, wsv, wsi);
        reduce_ws_kernel<<<BATCH, 256, 0, stream>>>(wsv, wsi, matches);
    } else {
        iou_partial_kernel<<<BATCH * NCHUNKS, 256, 0, stream>>>(
            anchors, gt, ious, nullptr, nullptr);
        reduce_full_kernel<<<BATCH, 256, 0, stream>>>(ious, matches);
    }
}